// CrossAttention_46462956208727
// MI455X (gfx1250) — compile-verified
//
#include <hip/hip_runtime.h>

// ---------------------------------------------------------------------------
// CrossAttention with a single global token collapses exactly:
//   K/V rows are identical per batch -> scores constant over m -> softmax is
//   uniform (1/N) -> attended[b,n,:] == v_b == g[b]@Wv+bv for all n.
//   out[b,n,:] = (g[b]@Wv + bv)@Wo + bo     (broadcast over n)
// Cost: two tiny f32 WMMA GEMMs + one 64 MiB broadcast store (BW-roofline
// ~2.8us at 23.3 TB/s). x, Wq, bq, Wk, bk are mathematically unused.
// ---------------------------------------------------------------------------

typedef __attribute__((ext_vector_type(2))) float v2f;
typedef __attribute__((ext_vector_type(4))) float f4;
typedef __attribute__((ext_vector_type(8))) float v8f;

#define BATCH      8
#define NPTS       4096
#define LOCAL_DIM  512
#define GLOBAL_DIM 128
#define HIDDEN_DIM 256

// grid=2 x 512 threads (16 wave32s each).
// Both blocks redundantly compute Y1 = Gpad@Wv + bv (tiny; Wv is L2-hot for
// the 2nd block), then split the 32 column tiles of Y2 = Y1@Wo + bo across
// blocks so the 512 KB Wo stream is spread over 2 WGPs with every wave owning
// exactly one WMMA tile. Full fp32 via V_WMMA_F32_16X16X4_F32.
__global__ __launch_bounds__(512) void ca_proj_wmma(
    const float* __restrict__ g,    // (8,128)
    const float* __restrict__ Wv,   // (128,256) row-major
    const float* __restrict__ bv,   // (256)
    const float* __restrict__ Wo,   // (256,512) row-major
    const float* __restrict__ bo,   // (512)
    float* __restrict__ y)          // (8,512) scratch out
{
    __shared__ float Gpad[16 * GLOBAL_DIM];   // zero-padded A for GEMM1 (8 KB)
    __shared__ float Y1[16 * HIDDEN_DIM];     // 16x256 intermediate (16 KB)

    const int tid  = threadIdx.x;
    const int wave = tid >> 5;      // 0..15
    const int lane = tid & 31;
    const int half = lane >> 4;     // selects K-pair within A/B operands
    const int l15  = lane & 15;     // row (A) / col (B,C,D) within tile

    // Stage Gpad: rows 0..7 = g, rows 8..15 = 0. Branch-free A reads after.
#pragma unroll
    for (int i = 0; i < 4; ++i) {
        const int idx = tid + i * 512;                 // 0..2047
        Gpad[idx] = (idx < BATCH * GLOBAL_DIM) ? g[idx] : 0.0f;
    }
    __syncthreads();

    // ---- GEMM1: Y1(16x256) = Gpad(16x128) @ Wv(128x256) + bv ---------------
    // wave w owns output columns [w*16, w*16+16); K pipelined in steps of 4.
    {
        const int ncol = wave * 16 + l15;
        const int m    = l15;
        const int kh   = half * 2;
        v8f c = {};
        // prefetch k0 = 0
        v2f a = *(const v2f*)&Gpad[m * GLOBAL_DIM + kh];
        v2f b;
        b.x = Wv[kh * HIDDEN_DIM + ncol];
        b.y = Wv[(kh + 1) * HIDDEN_DIM + ncol];
        for (int k0 = 0; k0 < GLOBAL_DIM - 4; k0 += 4) {
            const int kn = k0 + 4 + kh;
            v2f an = *(const v2f*)&Gpad[m * GLOBAL_DIM + kn];
            v2f bn;
            bn.x = Wv[kn * HIDDEN_DIM + ncol];
            bn.y = Wv[(kn + 1) * HIDDEN_DIM + ncol];
            c = __builtin_amdgcn_wmma_f32_16x16x4_f32(
                    false, a, false, b, (short)0, c, false, false);
            a = an; b = bn;
        }
        c = __builtin_amdgcn_wmma_f32_16x16x4_f32(
                false, a, false, b, (short)0, c, false, false);

        const float bias = bv[ncol];
#pragma unroll
        for (int j = 0; j < 8; ++j) {        // C/D: VGPR j -> row j + half*8
            const int row = j + half * 8;
            Y1[row * HIDDEN_DIM + ncol] = c[j] + bias;
        }
    }
    __syncthreads();

    // ---- GEMM2: Y2(16x512) = Y1(16x256) @ Wo(256x512) + bo -----------------
    // 32 column tiles; block b takes tiles [b*16, b*16+16), one per wave.
    {
        const int tile = blockIdx.x * 16 + wave;       // 0..31
        const int ncol = tile * 16 + l15;
        const int m    = l15;
        const int kh   = half * 2;
        v8f c = {};
        v2f a = *(const v2f*)&Y1[m * HIDDEN_DIM + kh];
        v2f b;
        b.x = Wo[kh * LOCAL_DIM + ncol];
        b.y = Wo[(kh + 1) * LOCAL_DIM + ncol];
        for (int k0 = 0; k0 < HIDDEN_DIM - 4; k0 += 4) {
            const int kn = k0 + 4 + kh;
            v2f an = *(const v2f*)&Y1[m * HIDDEN_DIM + kn];
            v2f bn;
            bn.x = Wo[kn * LOCAL_DIM + ncol];
            bn.y = Wo[(kn + 1) * LOCAL_DIM + ncol];
            c = __builtin_amdgcn_wmma_f32_16x16x4_f32(
                    false, a, false, b, (short)0, c, false, false);
            a = an; b = bn;
        }
        c = __builtin_amdgcn_wmma_f32_16x16x4_f32(
                false, a, false, b, (short)0, c, false, false);

        const float bias = bo[ncol];
        // Only rows 0..7 are real batches: VGPR j / lanes 0-15 -> row j.
        // Divergence is after this wave's last WMMA -> legal.
        if (half == 0) {
#pragma unroll
            for (int j = 0; j < 8; ++j)
                y[j * LOCAL_DIM + ncol] = c[j] + bias;
        }
    }
}

// Broadcast y[b][:] to out[b][n][:] for all n. Pure store-bandwidth kernel:
// 64 MiB of b128 non-temporal stores; the 16 KB source stays L2/L0 resident.
__global__ __launch_bounds__(256) void ca_broadcast(
    const float* __restrict__ y,    // (8,512)
    float* __restrict__ out)        // (8,4096,512)
{
    const long long total4 = (long long)BATCH * NPTS * LOCAL_DIM / 4; // 4,194,304
    const long long stride = (long long)gridDim.x * blockDim.x;
    const f4* y4 = (const f4*)y;
    f4* o4 = (f4*)out;
    for (long long i = (long long)blockIdx.x * blockDim.x + threadIdx.x;
         i < total4; i += stride) {
        const int row = (int)(i >> 7);       // 128 float4 per 512-float row
        const int b   = row >> 12;           // 4096 rows per batch
        const int c4  = (int)(i & 127);
        const f4 v = y4[b * (LOCAL_DIM / 4) + c4];
        __builtin_nontemporal_store(v, &o4[i]);
    }
}

extern "C" void kernel_launch(void* const* d_in, const int* in_sizes, int n_in,
                              void* d_out, int out_size, void* d_ws, size_t ws_size,
                              hipStream_t stream) {
    // setup_inputs order: x, g, Wq, bq, Wk, bk, Wv, bv, Wo, bo
    const float* g  = (const float*)d_in[1];
    const float* Wv = (const float*)d_in[6];
    const float* bv = (const float*)d_in[7];
    const float* Wo = (const float*)d_in[8];
    const float* bo = (const float*)d_in[9];
    float* y = (float*)d_ws;   // 8*512 floats = 16 KB scratch

    ca_proj_wmma<<<2, 512, 0, stream>>>(g, Wv, bv, Wo, bo, y);
    ca_broadcast<<<8192, 256, 0, stream>>>(y, (float*)d_out);
}